// UnitaryRNN_80547816669382
// MI455X (gfx1250) — compile-verified
//
#include <hip/hip_runtime.h>

// ---------------------------------------------------------------------------
// UnitaryRNN for MI455X (gfx1250, wave32)
//   Stage 1: batched expm of 32768 skew-symmetric 32x32 matrices using
//            V_WMMA_F32_16X16X4_F32 (FP32 matrix pipe -- precision needed for
//            the 512-step orthogonal recurrence).
//   Stage 2: sequential scan h <- M^T h, one wave per batch lane.
// ---------------------------------------------------------------------------

typedef __attribute__((ext_vector_type(2))) float v2f;
typedef __attribute__((ext_vector_type(8))) float v8f;

#define NDIM 32
#define LSTR 33                 // padded LDS row stride (bank-conflict free)
#define MATS_PER_BLOCK 4        // 4 waves/block = 128 threads
#define SEQ_LEN 512
#define BATCH 64
#define NMATS (SEQ_LEN * BATCH) // 32768
#define INPUT_SIZE 496

// ---------------------------------------------------------------------------
// One wave computes D(32x32) = alpha * (A x B) [+ I], A,B,D in LDS (stride 33).
// Fragment layouts per CDNA5 ISA 7.12.2:
//   A 16x4 f32 : lane l -> M=(l&15); VGPR0/1 = K = 4*kc + 2*(l>>4) + {0,1}
//   B 4x16 f32 : lane l -> N=(l&15); VGPR0/1 = rows K = 4*kc + 2*(l>>4) + {0,1}
//   C/D 16x16  : lane l -> N=(l&15); VGPR v -> M = v + 8*(l>>4)
// SCALE/ADDI are compile-time so the epilogue costs nothing when unused and
// alpha is always a literal (no runtime divides in the WMMA pipeline).
// ---------------------------------------------------------------------------
template <bool SCALE, bool ADDI>
__device__ __forceinline__ void wave_matmul32(const float* A, const float* B,
                                              float* D, int lane, float alpha)
{
    const int half = lane >> 4;   // 0 or 1
    const int l16  = lane & 15;

#pragma unroll
    for (int mt = 0; mt < 2; ++mt) {
        // Load all A fragments for this 16-row tile (reused across nt)
        v2f afrag[8];
        const int m = mt * 16 + l16;
#pragma unroll
        for (int kc = 0; kc < 8; ++kc) {
            const int k = kc * 4 + half * 2;
            afrag[kc][0] = A[m * LSTR + k];
            afrag[kc][1] = A[m * LSTR + k + 1];
        }
#pragma unroll
        for (int nt = 0; nt < 2; ++nt) {
            const int n = nt * 16 + l16;
            v8f c = {0.f, 0.f, 0.f, 0.f, 0.f, 0.f, 0.f, 0.f};
#pragma unroll
            for (int kc = 0; kc < 8; ++kc) {
                const int k = kc * 4 + half * 2;
                v2f b;
                b[0] = B[(k    ) * LSTR + n];
                b[1] = B[(k + 1) * LSTR + n];
                // D = A x B + C, fp32 matrix pipe
                c = __builtin_amdgcn_wmma_f32_16x16x4_f32(
                        /*neg_a=*/false, afrag[kc],
                        /*neg_b=*/false, b,
                        /*c_mod=*/(short)0, c,
                        /*reuse_a=*/false, /*reuse_b=*/false);
            }
#pragma unroll
            for (int v = 0; v < 8; ++v) {
                const int mm = mt * 16 + half * 8 + v;
                float val = c[v];
                if (SCALE) val *= alpha;
                if (ADDI && mm == n) val += 1.0f;
                D[mm * LSTR + n] = val;
            }
        }
    }
}

// ---------------------------------------------------------------------------
// Stage 1: expm via scaling-and-squaring + order-8 Taylor (Horner).
// One wave per matrix; 4 matrices per block.
// ---------------------------------------------------------------------------
__global__ __launch_bounds__(MATS_PER_BLOCK * 32)
void expm_kernel(const float* __restrict__ text, float* __restrict__ ws)
{
    __shared__ float lds[MATS_PER_BLOCK * 3 * (NDIM * LSTR)];

    const int wave = threadIdx.x >> 5;
    const int lane = threadIdx.x & 31;
    const int g    = blockIdx.x * MATS_PER_BLOCK + wave;  // (t*64 + b)

    float* Bm = lds + wave * 3 * (NDIM * LSTR);
    float* T  = Bm + NDIM * LSTR;
    float* D  = T  + NDIM * LSTR;

    // ---- gather skew-symmetric matrix into T (lane = row i) --------------
    {
        const int    i    = lane;
        const size_t base = (size_t)g * INPUT_SIZE;
        T[i * LSTR + i] = 0.0f;
        for (int j = i + 1; j < NDIM; ++j) {
            // ix(i,j) - 1 = i*(2n-i-3)/2 + j - 1,  n = 32
            const float v = text[base + (i * (61 - i)) / 2 + j - 1];
            T[i * LSTR + j] =  v;
            T[j * LSTR + i] = -v;
        }
    }
    __syncthreads();   // cross-lane LDS visibility before norm/scale

    // ---- infinity norm -> scaling exponent s (wave-uniform) --------------
    float rs = 0.0f;
#pragma unroll
    for (int j = 0; j < NDIM; ++j) rs += fabsf(T[lane * LSTR + j]);
#pragma unroll
    for (int off = 16; off > 0; off >>= 1)
        rs = fmaxf(rs, __shfl_xor(rs, off, 32));

    int s = 0;
    if (rs > 0.5f) s = (int)ceilf(log2f(rs)) + 1;   // => ||A/2^s|| <= 0.5
    s = (s < 0) ? 0 : (s > 30 ? 30 : s);
    const float sc = exp2f(-(float)s);

    // ---- Bm = A * 2^-s ; T = I + Bm/8 (innermost Horner term) ------------
    {
        const int i = lane;
#pragma unroll
        for (int j = 0; j < NDIM; ++j) {
            const float a = T[i * LSTR + j] * sc;
            Bm[i * LSTR + j] = a;
            T[i * LSTR + j]  = a * 0.125f + ((i == j) ? 1.0f : 0.0f);
        }
    }

    // ---- Horner: T = I + (Bm/k) * T, k = 7..1  (7 matmuls) ---------------
    // Fully unrolled so alpha = 1/k is a literal (no runtime fp32 divide).
#pragma unroll
    for (int k = 7; k >= 1; --k) {
        wave_matmul32<true, true>(Bm, T, D, lane, 1.0f / (float)k);
        float* t = T; T = D; D = t;
    }
    // ---- s squarings: T = T * T (dynamic, wave-uniform) ------------------
    for (int q = 0; q < s; ++q) {
        wave_matmul32<false, false>(T, T, D, lane, 1.0f);
        float* t = T; T = D; D = t;
    }

    // ---- store exp(A) row-major to workspace (coalesced per row) ---------
    float* dst = ws + (size_t)g * (NDIM * NDIM);
#pragma unroll
    for (int r = 0; r < NDIM; ++r)
        dst[r * NDIM + lane] = T[r * LSTR + lane];
}

// ---------------------------------------------------------------------------
// Stage 2: scan. One wave per batch element. h_new[j] = sum_i h[i]*M[i][j].
// Lane j owns column j; broadcast h[i] via shuffle. Prefetch next matrix.
// ---------------------------------------------------------------------------
__global__ __launch_bounds__(32)
void scan_kernel(const float* __restrict__ M, float* __restrict__ out)
{
    const int b    = blockIdx.x;   // 0..63
    const int lane = threadIdx.x;  // 0..31

    float r = (lane == 0) ? 1.0f : 0.0f;

    for (int t = 0; t < SEQ_LEN; ++t) {
        const float* Mp = M + ((size_t)t * BATCH + b) * (NDIM * NDIM);

        if (t + 1 < SEQ_LEN) {
            // warm next matrix (4KB) into cache: 32 lanes x 128B
            const float* Mn = Mp + (size_t)BATCH * (NDIM * NDIM);
            __builtin_prefetch(Mn + lane * 32, 0, 1);
        }

        float col[NDIM];
#pragma unroll
        for (int i = 0; i < NDIM; ++i) col[i] = Mp[i * NDIM + lane];

        float acc = 0.0f;
#pragma unroll
        for (int i = 0; i < NDIM; ++i)
            acc = fmaf(__shfl(r, i, 32), col[i], acc);
        r = acc;

        out[((size_t)t * BATCH + b) * NDIM + lane] = r;
    }
}

// ---------------------------------------------------------------------------
extern "C" void kernel_launch(void* const* d_in, const int* in_sizes, int n_in,
                              void* d_out, int out_size, void* d_ws, size_t ws_size,
                              hipStream_t stream)
{
    (void)in_sizes; (void)n_in; (void)out_size; (void)ws_size;

    const float* text = (const float*)d_in[0];   // (512, 64, 496) f32
    float*       ws   = (float*)d_ws;            // needs 32768*1024*4 = 128 MiB
    float*       out  = (float*)d_out;           // (512, 64, 32) f32

    expm_kernel<<<NMATS / MATS_PER_BLOCK, MATS_PER_BLOCK * 32, 0, stream>>>(text, ws);
    scan_kernel<<<BATCH, 32, 0, stream>>>(ws, out);
}